// get_loss_84610855731464
// MI455X (gfx1250) — compile-verified
//
#include <hip/hip_runtime.h>

// ---------------- problem constants (match reference) ----------------
#define N_NODES   1500
#define NUM_OBJ   160
#define NUM_PRED  27
#define E_TOTAL   2248500LL   // N*N - N, even
// ALPHA = BETA = 1.0, GAMMA = 2.0 (t*t)

typedef float v2f __attribute__((ext_vector_type(2)));
typedef float v8f __attribute__((ext_vector_type(8)));

// sigmoid via v_rcp_f32 (1 ulp) instead of full IEEE divide sequence;
// denominator is in [1, +inf), well-conditioned for rcp.
__device__ __forceinline__ float sigmoidf_(float x) {
  return __builtin_amdgcn_rcpf(1.0f + expf(-x));
}

// focal term given probability p already gathered and alpha
__device__ __forceinline__ float focalf_(float p, float a) {
  float t = 1.0f - p;                 // (1-p)^2 == pow(1-p, 2.0) for all finite p
  return -a * t * t * logf(p);
}

// rare path: row has at least one relation (multi-hot possible)
__device__ __noinline__ float edge_row_loss_slow(unsigned m,
    const float* __restrict__ row,
    const float* __restrict__ s_alpha,
    float p0 /* sigmoid of column 0, already computed */)
{
  int id = __builtin_ctz(m);          // argmax of one-hot = lowest set bit
  float probs = 0.0f;
  if (m & 1u) { probs = p0; m &= ~1u; }
  while (m) {                         // sum of sigmoids over remaining hot columns
    int c = __builtin_ctz(m);
    m &= (m - 1u);
    probs += sigmoidf_(row[c]);
  }
  return focalf_(probs, s_alpha[id]);
}

// ---------------- kernel 1: zero the E-row bitmask (uint4 stores) -----------
__global__ void zero_mask_kernel(uint4* __restrict__ mask4, int n4) {
  int i = blockIdx.x * blockDim.x + threadIdx.x;
  if (i < n4) mask4[i] = make_uint4(0u, 0u, 0u, 0u);
}

// ---------------- kernel 2: scatter relations into the bitmask --------------
__global__ void scatter_kernel(const int* __restrict__ rel_ij,
                               const int* __restrict__ rel_c,
                               unsigned* __restrict__ mask, int M) {
  int m = blockIdx.x * blockDim.x + threadIdx.x;
  if (m >= M) return;
  int i = rel_ij[2 * m];
  int j = rel_ij[2 * m + 1];
  if (i == j) return;                                  // dropped rows
  long long row = (long long)i * (N_NODES - 1) + j - (i < j ? 1 : 0);
  atomicOr(&mask[row], 1u << rel_c[m]);                // idempotent -> deterministic
}

// ---------------- kernel 3: edge focal loss, WMMA f32 accumulation ----------
__global__ void __launch_bounds__(256)
edge_loss_kernel(const float* __restrict__ edge,
                 const float* __restrict__ pred_alpha,
                 const unsigned* __restrict__ mask,
                 double* __restrict__ partials) {
  __shared__ float  s_alpha[NUM_PRED];
  __shared__ double s_wave[8];
  if (threadIdx.x < NUM_PRED) s_alpha[threadIdx.x] = pred_alpha[threadIdx.x];
  __syncthreads();
  const float alpha0 = s_alpha[0];    // register-cache the hot alpha

  long long tid      = (long long)blockIdx.x * blockDim.x + threadIdx.x;
  long long nthreads = (long long)gridDim.x * blockDim.x;
  long long stride   = nthreads * 2;                       // 2 rows per thread/iter
  long long niter    = (E_TOTAL + stride - 1) / stride;    // uniform -> EXEC all-1s

  v8f acc = {};                       // 16x16 f32 accumulator (8 VGPRs)
  v2f ones; ones.x = 1.0f; ones.y = 1.0f;  // B = ones(4x16): D = A*1 + C

  for (long long it = 0; it < niter; ++it) {
    long long base = tid * 2 + it * stride;
    v2f a; a.x = 0.0f; a.y = 0.0f;    // A = 16x4 f32 tile: 2 loss values/lane
    if (base < E_TOTAL) {             // base even, E even -> base+1 in range
      // ---- issue ALL loads for this pair up front (one wait, 3 in flight) --
      uint2 mp = *(const uint2*)(mask + base);          // masks for both rows
      const float* row0 = edge + base * (long long)NUM_PRED;
      const float* row1 = row0 + NUM_PRED;
      float x0 = row0[0];                               // column-0 logits,
      float x1 = row1[0];                               // needed in ~96% of rows
      // sigmoid of column 0 computed unconditionally (reused by slow path)
      float p0 = sigmoidf_(x0);
      float p1 = sigmoidf_(x1);
      a.x = (mp.x == 0u) ? focalf_(p0, alpha0)
                         : edge_row_loss_slow(mp.x, row0, s_alpha, p0);
      a.y = (mp.y == 0u) ? focalf_(p1, alpha0)
                         : edge_row_loss_slow(mp.y, row1, s_alpha, p1);
    }
    // exact f32 FMA reduction on the matrix unit:
    // acc[m][n] += sum_k A[m][k]  (every column n identical)
    acc = __builtin_amdgcn_wmma_f32_16x16x4_f32(false, a, false, ones,
                                                (short)0, acc, false, false);
  }

  // column N=0 of D lives in lane 0 (M=0..7) and lane 16 (M=8..15)
  double wsum = 0.0;
#pragma unroll
  for (int v = 0; v < 8; ++v) {
    int b = __float_as_int(acc[v]);
    wsum += (double)__int_as_float(__builtin_amdgcn_readlane(b, 0));
    wsum += (double)__int_as_float(__builtin_amdgcn_readlane(b, 16));
  }
  if ((threadIdx.x & 31) == 0) s_wave[threadIdx.x >> 5] = wsum;
  __syncthreads();
  if (threadIdx.x == 0) {
    double s = 0.0;
    int nw = blockDim.x >> 5;
    for (int w = 0; w < nw; ++w) s += s_wave[w];   // fixed order
    partials[blockIdx.x] = s;
  }
}

// ---------------- kernel 4: object focal loss (tiny gather) -----------------
__global__ void obj_loss_kernel(const float* __restrict__ node,
                                const float* __restrict__ obj_alpha,
                                const int* __restrict__ gt,
                                double* __restrict__ partials) {
  __shared__ double s_red[256];
  int tid   = blockIdx.x * blockDim.x + threadIdx.x;
  int total = gridDim.x * blockDim.x;
  double s = 0.0;
  for (int i = tid; i < N_NODES; i += total) {
    int g = gt[i];
    float x = node[(long long)i * NUM_OBJ + g];
    s += (double)focalf_(sigmoidf_(x), obj_alpha[g]);
  }
  s_red[threadIdx.x] = s;
  __syncthreads();
  for (int off = blockDim.x >> 1; off > 0; off >>= 1) {
    if ((int)threadIdx.x < off) s_red[threadIdx.x] += s_red[threadIdx.x + off];
    __syncthreads();
  }
  if (threadIdx.x == 0) partials[blockIdx.x] = s_red[0];
}

// ---------------- kernel 5: deterministic final combine ---------------------
__global__ void finalize_kernel(const double* __restrict__ ep, int ne,
                                const double* __restrict__ op, int no,
                                float* __restrict__ out) {
  __shared__ double s_red[256];
  double s = 0.0;
  for (int i = threadIdx.x; i < ne; i += blockDim.x) s += ep[i];  // fixed strides
  s_red[threadIdx.x] = s;
  __syncthreads();
  for (int off = blockDim.x >> 1; off > 0; off >>= 1) {
    if ((int)threadIdx.x < off) s_red[threadIdx.x] += s_red[threadIdx.x + off];
    __syncthreads();
  }
  if (threadIdx.x == 0) {
    double es = s_red[0];
    double os = 0.0;
    for (int i = 0; i < no; ++i) os += op[i];
    double obj_mean  = os / (double)N_NODES;
    double pred_mean = es / (double)E_TOTAL;
    out[0] = (float)(obj_mean + pred_mean);   // ALPHA=BETA=1
  }
}

// ---------------- launcher --------------------------------------------------
extern "C" void kernel_launch(void* const* d_in, const int* in_sizes, int n_in,
                              void* d_out, int out_size, void* d_ws, size_t ws_size,
                              hipStream_t stream) {
  const float* node       = (const float*)d_in[0];   // (N, 160)
  const float* edge       = (const float*)d_in[1];   // (E, 27)
  const float* obj_alpha  = (const float*)d_in[2];   // (160,)
  const float* pred_alpha = (const float*)d_in[3];   // (27,)
  const int*   gt         = (const int*)d_in[4];     // (N,)
  const int*   rel_ij     = (const int*)d_in[5];     // (M, 2)
  const int*   rel_c      = (const int*)d_in[6];     // (M,)
  int M = in_sizes[5] / 2;

  // workspace layout: [ E x u32 masks | GE doubles | GO doubles ]
  unsigned* mask = (unsigned*)d_ws;
  size_t mask_bytes = (size_t)E_TOTAL * sizeof(unsigned);   // 8,994,000 B (16-aligned)
  double* ep = (double*)((char*)d_ws + mask_bytes);
  const int GE = 1024;
  double* op = ep + GE;
  const int GO = 8;

  int n4 = (int)(E_TOTAL / 4);   // E divisible by 4
  zero_mask_kernel<<<(n4 + 255) / 256, 256, 0, stream>>>((uint4*)mask, n4);
  scatter_kernel<<<(M + 255) / 256, 256, 0, stream>>>(rel_ij, rel_c, mask, M);
  edge_loss_kernel<<<GE, 256, 0, stream>>>(edge, pred_alpha, mask, ep);
  obj_loss_kernel<<<GO, 256, 0, stream>>>(node, obj_alpha, gt, op);
  finalize_kernel<<<1, 256, 0, stream>>>(ep, GE, op, GO, (float*)d_out);
}